// GC_withres_52613349376871
// MI455X (gfx1250) — compile-verified
//
#include <hip/hip_runtime.h>

#define NN 100000
#define EE 1600000
#define DD 128

typedef __attribute__((ext_vector_type(2))) float v2f;
typedef __attribute__((ext_vector_type(8))) float v8f;

// ---------------------------------------------------------------------------
// Kernel 1: zero the workspace (acc[N*128] and deg[N]) every call.
// ---------------------------------------------------------------------------
__global__ void zero_ws_kernel(float* __restrict__ acc, float* __restrict__ deg) {
    size_t i      = (size_t)blockIdx.x * blockDim.x + threadIdx.x;
    size_t stride = (size_t)gridDim.x * blockDim.x;
    const size_t total = (size_t)NN * DD;
    for (size_t j = i; j < total; j += stride) acc[j] = 0.0f;
    for (size_t j = i; j < NN;    j += stride) deg[j] = 0.0f;
}

// ---------------------------------------------------------------------------
// Kernel 2: support = X @ W via V_WMMA_F32_16X16X4_F32 (fp32, exact).
// One block = 256 threads = 8 wave32s, handles a 16-row strip of X.
// Wave w computes the 16x16 tile at columns [16w, 16w+16).
// 16x4 fp32 A fragment: lanes 0-15 rows M=0..15; VGPR0 holds K = 2*half,
// VGPR1 holds K = 2*half+1 (half = lane>=16). B symmetric.
// ---------------------------------------------------------------------------
__global__ void __launch_bounds__(256)
gemm_wmma_kernel(const float* __restrict__ X, const float* __restrict__ W,
                 float* __restrict__ support) {
    __shared__ float sX[16 * DD];  // 8 KB strip of X

    const int tid = threadIdx.x;
    const int m0  = blockIdx.x * 16;

    // Cooperative stage of the 16x128 X strip into LDS (512 float4 / 256 thr).
    const float4* src = (const float4*)(X + (size_t)m0 * DD);
    float4*       dst = (float4*)sX;
    dst[tid]       = src[tid];
    dst[tid + 256] = src[tid + 256];
    __syncthreads();

    const int wave = tid >> 5;
    const int lane = tid & 31;
    const int half = lane >> 4;   // 0: lanes 0-15, 1: lanes 16-31
    const int lr   = lane & 15;
    const int n0   = wave * 16;

    v8f c = {};
#pragma unroll
    for (int k = 0; k < DD; k += 4) {
        // A fragment from LDS: float2 at (row=lr, cols k+2*half .. +1)
        v2f a = *(const v2f*)(&sX[lr * DD + k + 2 * half]);
        // B fragment from global (W is row-major [128,128], hot in L2)
        v2f b;
        b.x = W[(size_t)(k + 2 * half)     * DD + n0 + lr];
        b.y = W[(size_t)(k + 2 * half + 1) * DD + n0 + lr];
        c = __builtin_amdgcn_wmma_f32_16x16x4_f32(
            /*neg_a=*/false, a, /*neg_b=*/false, b,
            /*c_mod=*/(short)0, c, /*reuse_a=*/false, /*reuse_b=*/false);
    }

    // C/D layout: VGPR r -> M = r + 8*half, N = lr
#pragma unroll
    for (int r = 0; r < 8; ++r) {
        int row = m0 + r + 8 * half;
        support[(size_t)row * DD + n0 + lr] = c[r];
    }
}

// ---------------------------------------------------------------------------
// Kernel 3: deg[c] = segment_sum(vals by cols) via fp32 atomics (L2-resident).
// ---------------------------------------------------------------------------
__global__ void degree_kernel(const float* __restrict__ vals,
                              const int* __restrict__ cols,
                              float* __restrict__ deg) {
    int i      = blockIdx.x * blockDim.x + threadIdx.x;
    int stride = gridDim.x * blockDim.x;
    for (int e = i; e < EE; e += stride)
        atomicAdd(&deg[cols[e]], vals[e]);
}

// ---------------------------------------------------------------------------
// Kernel 4: SpMM scatter. One wave32 per edge; each lane owns one float4
// chunk of the 128-wide feature row. Gathers hit the 192 MB L2 (support is
// 51 MB), atomics resolve in L2 atomic units.
//   acc[rows[e]] += vals[e]/deg[cols[e]] * support[cols[e]]
// ---------------------------------------------------------------------------
__global__ void __launch_bounds__(256)
spmm_kernel(const float* __restrict__ vals, const int* __restrict__ rows,
            const int* __restrict__ cols, const float* __restrict__ deg,
            const float* __restrict__ support, float* __restrict__ acc) {
    int e = blockIdx.x * 8 + (threadIdx.x >> 5);
    if (e >= EE) return;
    int lane = threadIdx.x & 31;

    int   r    = rows[e];
    int   cidx = cols[e];
    float coef = vals[e] / deg[cidx];

    float4 sv = ((const float4*)(support + (size_t)cidx * DD))[lane];
    float* d  = acc + (size_t)r * DD + lane * 4;
    atomicAdd(d + 0, coef * sv.x);
    atomicAdd(d + 1, coef * sv.y);
    atomicAdd(d + 2, coef * sv.z);
    atomicAdd(d + 3, coef * sv.w);
}

// ---------------------------------------------------------------------------
// Kernel 5: out = (support + 0.5*(0.5*support + 0.5*acc)) / 1.5 + bias
//              = (1.25*support + 0.25*acc) * (1/1.5) + bias       (in place)
// ---------------------------------------------------------------------------
__global__ void final_kernel(const float* __restrict__ acc,
                             const float* __restrict__ bias,
                             float* __restrict__ out) {
    size_t i      = (size_t)blockIdx.x * blockDim.x + threadIdx.x;
    size_t stride = (size_t)gridDim.x * blockDim.x;
    const size_t total = (size_t)NN * DD;
    const float inv = 1.0f / 1.5f;
    for (size_t j = i; j < total; j += stride) {
        float sup = out[j];
        out[j] = (1.25f * sup + 0.25f * acc[j]) * inv + bias[j & (DD - 1)];
    }
}

// ---------------------------------------------------------------------------
extern "C" void kernel_launch(void* const* d_in, const int* in_sizes, int n_in,
                              void* d_out, int out_size, void* d_ws, size_t ws_size,
                              hipStream_t stream) {
    const float* x    = (const float*)d_in[0];  // [N, 128]
    const float* w    = (const float*)d_in[1];  // [128, 128]
    const float* bias = (const float*)d_in[2];  // [128]
    const float* vals = (const float*)d_in[3];  // [E]
    const int*   rows = (const int*)d_in[4];    // [E]
    const int*   cols = (const int*)d_in[5];    // [E]
    float*       out  = (float*)d_out;          // [N, 128] (also holds support)

    float* acc = (float*)d_ws;                  // N*128 floats
    float* deg = acc + (size_t)NN * DD;         // N floats

    zero_ws_kernel <<<2048, 256, 0, stream>>>(acc, deg);
    gemm_wmma_kernel<<<NN / 16, 256, 0, stream>>>(x, w, out);       // 6250 blocks
    degree_kernel  <<<2048, 256, 0, stream>>>(vals, cols, deg);
    spmm_kernel    <<<EE / 8, 256, 0, stream>>>(vals, rows, cols, deg, out, acc);
    final_kernel   <<<4096, 256, 0, stream>>>(acc, bias, out);
}